// InitLayer_24240795419358
// MI455X (gfx1250) — compile-verified
//
#include <hip/hip_runtime.h>

typedef __attribute__((ext_vector_type(2))) float v2f;
typedef __attribute__((ext_vector_type(4))) float v4f;
typedef __attribute__((ext_vector_type(8))) float v8f;

#define N_ROWS 2048
#define G_DIM  512
#define F_DIM  128

// ---------------------------------------------------------------------------
// Kernel A: dW = densities @ weight   [2048,512] x [512,128] -> [2048,128]
// One wave32 per 16x16 output tile; K-loop in steps of 4 with
// V_WMMA_F32_16X16X4_F32 (full fp32 precision; GEMM is ~0.27 GFLOP, trivial).
// 1024 tiles -> 1024 waves -> 256 blocks of 128 threads.
// ---------------------------------------------------------------------------
__global__ __launch_bounds__(128) void gemm_dw_kernel(
    const float* __restrict__ D, const float* __restrict__ W,
    float* __restrict__ dW)
{
    const int wave  = (blockIdx.x * blockDim.x + threadIdx.x) >> 5;
    const int lane  = threadIdx.x & 31;
    const int mTile = wave >> 3;            // 128 row tiles
    const int nTile = wave & 7;             // 8 col tiles
    const int half  = lane >> 4;            // 0 -> K=0,1 ; 1 -> K=2,3
    const int l     = lane & 15;
    const int m     = mTile * 16 + l;       // A row for this lane
    const int n     = nTile * 16 + l;       // B/C col for this lane

    const float* Drow = D + m * G_DIM + 2 * half;      // contiguous pair per lane
    const float* Wcol = W + 2 * half * F_DIM + n;      // strided pair per lane

    v8f acc = {};
#pragma unroll 4
    for (int k = 0; k < G_DIM; k += 4) {
        v2f a = *(const v2f*)(Drow + k);               // A: K=k+2h, k+2h+1
        v2f b;
        b.x = Wcol[k * F_DIM];                         // B: K=k+2h
        b.y = Wcol[k * F_DIM + F_DIM];                 // B: K=k+2h+1
        acc = __builtin_amdgcn_wmma_f32_16x16x4_f32(
            /*neg_a=*/false, a, /*neg_b=*/false, b,
            /*c_mod=*/(short)0, acc, /*reuse_a=*/false, /*reuse_b=*/false);
    }

    // C/D layout: VGPR v -> M = v + 8*half (lanes 16-31 offset rows by 8), N = l
#pragma unroll
    for (int v = 0; v < 8; ++v) {
        dW[(mTile * 16 + half * 8 + v) * F_DIM + n] = acc[v];
    }
}

// ---------------------------------------------------------------------------
// Kernel B: out[n,i,k] = D[n,i] * dW[n,k]  — pure streaming, 512 MiB of
// writes; this is the whole runtime (~23 us at 23.3 TB/s). Each block: one n,
// 32 i values. Each thread loads one float4 of dW[n] once, reuses it for 4 i
// rows, non-temporal B128 stores, fully coalesced.
// ---------------------------------------------------------------------------
__global__ __launch_bounds__(256) void expand_kernel(
    const float* __restrict__ D, const float* __restrict__ dW,
    float* __restrict__ out)
{
    const int n  = blockIdx.x >> 4;              // 2048
    const int ic = (blockIdx.x & 15) * 32;       // i-chunk base (32 i per block)
    const int k4 = (threadIdx.x & 31) * 4;       // 32 lanes cover 128 floats
    const int ig = (threadIdx.x >> 5) * 4;       // 8 groups x 4 i each

    v4f w = *(const v4f*)(dW + n * F_DIM + k4);
    const float* drow = D + n * G_DIM + ic + ig;
    float* o = out + ((size_t)n * G_DIM + (size_t)(ic + ig)) * F_DIM + k4;

#pragma unroll
    for (int j = 0; j < 4; ++j) {
        float s = drow[j];
        v4f r = w * s;
        __builtin_nontemporal_store(r, (v4f*)(o + (size_t)j * F_DIM));
    }
}

extern "C" void kernel_launch(void* const* d_in, const int* in_sizes, int n_in,
                              void* d_out, int out_size, void* d_ws, size_t ws_size,
                              hipStream_t stream)
{
    const float* D = (const float*)d_in[0];   // densities [2048, 512]
    const float* W = (const float*)d_in[1];   // weight    [512, 128]
    float* out = (float*)d_out;               // [2048, 512, 128]
    float* dW  = (float*)d_ws;                // scratch: 2048*128*4 = 1 MiB

    gemm_dw_kernel<<<256, 128, 0, stream>>>(D, W, dW);
    expand_kernel<<<N_ROWS * (G_DIM / 32), 256, 0, stream>>>(D, dW, out);
}